// LaplaceNet_283467842453
// MI455X (gfx1250) — compile-verified
//
#include <hip/hip_runtime.h>

// ---------------------------------------------------------------------------
// LaplaceNet on gfx1250 (MI455X): 4096x4096 fp32 Jacobi with boundary costs.
//   K1 init      : V0, objects_bounds, bit-packed interior mask (ballot_w32)
//   K2 hconv     : horizontal 11-tap sum as banded GEMM via V_WMMA_F32_16X16X4_F32
//   K3 vconv     : vertical 11-tap sum, build fused B field, emit C_pos output
//   K4 jacobi x20: Vn = interior ? 0.25*(N+S+E+W) + B : B   (edge replicate)
//   K5 finalize  : clamp >=1.0 -> 0.95 on interior cells, write d_out
// K3/K4/K5 process 4 cells/thread with float4 (b128) traffic.
// ---------------------------------------------------------------------------

#define RES 4096
#define NCELLS (RES * RES)          // 16,777,216
#define TILES_X 256                 // RES / 16

typedef __attribute__((ext_vector_type(2))) float v2f;
typedef __attribute__((ext_vector_type(8))) float v8f;

// ---------------------------------------------------------------- K1: init
__global__ void k_init(const float* __restrict__ x,
                       const float* __restrict__ btf,
                       const float* __restrict__ bc,
                       float* __restrict__ V0,
                       float* __restrict__ ob,
                       unsigned* __restrict__ mask) {
  int idx = blockIdx.x * blockDim.x + threadIdx.x;
  float bt = btf[idx];
  float b  = bc[idx];
  bool obj = bt > 0.0f;
  V0[idx] = obj ? b : x[idx];
  // objects_bounds = bc * (bt==1) * (bc>0)
  ob[idx] = (bt == 1.0f && b > 0.0f) ? b : 0.0f;
  // interior bit-mask, one dword per 32 consecutive cells of a row
  unsigned m = __builtin_amdgcn_ballot_w32(!obj);
  if ((threadIdx.x & 31) == 0) mask[idx >> 5] = m;
}

// ------------------------------------------- K2: horizontal conv via WMMA
// Each wave produces one 16x16 tile of H, H[i][n] = sum_{d=-5..5} ob[i][n+d]
// (zero padding), expressed as 16x48 x 48x16 banded matmul:
//   12 chained V_WMMA_F32_16X16X4_F32 over K-window cols [base-16, base+32).
__global__ void k_hconv_wmma(const float* __restrict__ ob,
                             float* __restrict__ H) {
  const int lane = threadIdx.x & 31;
  const int wave = threadIdx.x >> 5;        // 4 waves / block
  const int tile = blockIdx.x * 4 + wave;
  const int tr = tile >> 8;                 // tile row 0..255
  const int tc = tile & 255;                // tile col 0..255
  const int row0 = tr * 16;
  const int base = tc * 16;                 // first output column

  const int LDW = 66;                       // padded LDS row stride
  __shared__ float smem[4][16 * 66];
  float* s = smem[wave];

  // Cooperative load: 16 rows x 64 cols window, cols [base-16, base+48),
  // zero-padded outside [0, RES).  Power-of-two split: per unrolled step t,
  // row = t/2 and col = (t&1)*32 + lane are compile-time affine -> no idiv.
  // Only cols [0,48) of the window are consumed by the K=48 WMMA chain.
  #pragma unroll
  for (int t = 0; t < 32; ++t) {
    const int r = t >> 1;
    const int c = ((t & 1) << 5) + lane;
    int gc = base - 16 + c;
    float v = 0.0f;
    if (gc >= 0 && gc < RES) v = ob[(row0 + r) * RES + gc];
    s[r * LDW + c] = v;
  }
  __syncthreads();

  const int half = lane >> 4;               // 0: lanes 0-15, 1: lanes 16-31
  const int mn   = lane & 15;               // m for A, n for B and D

  v8f acc = {};
  #pragma unroll
  for (int q = 0; q < 12; ++q) {
    // A (16x4 f32): lane holds row mn, K pair starting at kb (8B aligned)
    int kb = q * 4 + half * 2;
    v2f a;
    a.x = s[mn * LDW + kb];
    a.y = s[mn * LDW + kb + 1];
    // B (4x16 f32): constant band weights. Input col offset from base = k-16;
    // tap contributes iff |(k-16) - n| <= 5.
    int c0 = kb - 16 - mn;
    v2f b;
    b.x = (c0 >= -5 && c0 <= 5) ? 1.0f : 0.0f;
    b.y = (c0 + 1 >= -5 && c0 + 1 <= 5) ? 1.0f : 0.0f;
    acc = __builtin_amdgcn_wmma_f32_16x16x4_f32(
        /*neg_a=*/false, a, /*neg_b=*/false, b,
        /*c_mod=*/(short)0, acc, /*reuse_a=*/false, /*reuse_b=*/false);
  }

  // D layout: VGPR r -> rows r (lanes 0-15) / 8+r (lanes 16-31), col = lane&15
  const int col = base + mn;
  const int mbase = half * 8;
  #pragma unroll
  for (int r = 0; r < 8; ++r) {
    H[(row0 + mbase + r) * RES + col] = acc[r];
  }
}

// ---------------------------- K3: vertical conv + fused cost/boundary field
__global__ void k_vconv_buildB(const float* __restrict__ H,
                               const float* __restrict__ btf,
                               const float* __restrict__ bc,
                               const float* __restrict__ C,
                               float* __restrict__ B,
                               float* __restrict__ cpos_out) {
  int t = blockIdx.x * blockDim.x + threadIdx.x;   // 4 cells / thread
  int c0 = t << 2;
  int i = c0 >> 12;                                // row
  int j = c0 & (RES - 1);                          // first col of quad
  float4 sum = make_float4(0.f, 0.f, 0.f, 0.f);
  #pragma unroll
  for (int d = -5; d <= 5; ++d) {
    int r = i + d;
    if (r >= 0 && r < RES) {
      float4 h = *(const float4*)(H + (size_t)r * RES + j);
      sum.x += h.x; sum.y += h.y; sum.z += h.z; sum.w += h.w;
    }
  }
  const float inv = 1.0f / 121.0f;                 // full 1/(11*11) once
  float4 bt = *(const float4*)(btf + c0);
  float4 cv = *(const float4*)(C + c0);
  float4 bv = *(const float4*)(bc + c0);
  float4 cp = make_float4(fmaxf(cv.x, 0.f), fmaxf(cv.y, 0.f),
                          fmaxf(cv.z, 0.f), fmaxf(cv.w, 0.f));
  float4 Bv;
  Bv.x = (bt.x == 0.f) ? (cp.x + sum.x * inv) : bv.x;
  Bv.y = (bt.y == 0.f) ? (cp.y + sum.y * inv) : bv.y;
  Bv.z = (bt.z == 0.f) ? (cp.z + sum.z * inv) : bv.z;
  Bv.w = (bt.w == 0.f) ? (cp.w + sum.w * inv) : bv.w;
  *(float4*)(B + c0) = Bv;
  *(float4*)(cpos_out + c0) = cp;                  // second tuple output
}

// ------------------------------------------------------- K4: Jacobi sweep
__global__ void k_jacobi(const float* __restrict__ Vin,
                         const float* __restrict__ B,
                         const unsigned* __restrict__ mask,
                         float* __restrict__ Vout) {
  int t = blockIdx.x * blockDim.x + threadIdx.x;   // 4 cells / thread
  int c0 = t << 2;
  int i = c0 >> 12;
  int j0 = c0 & (RES - 1);
  const int iu = (i > 0) ? i - 1 : 0;              // edge replicate rows
  const int id = (i < RES - 1) ? i + 1 : RES - 1;
  const float* rowC = Vin + (size_t)i * RES;
  float4 cc = *(const float4*)(rowC + j0);
  float4 up = *(const float4*)(Vin + (size_t)iu * RES + j0);
  float4 dn = *(const float4*)(Vin + (size_t)id * RES + j0);
  float lf = (j0 > 0)        ? rowC[j0 - 1] : cc.x;  // edge replicate cols
  float rt = (j0 + 4 < RES)  ? rowC[j0 + 4] : cc.w;
  float4 b = *(const float4*)(B + c0);
  unsigned mw = mask[c0 >> 5] >> (c0 & 31);
  float4 v;
  v.x = 0.25f * (up.x + dn.x + lf   + cc.y) + b.x;
  v.y = 0.25f * (up.y + dn.y + cc.x + cc.z) + b.y;
  v.z = 0.25f * (up.z + dn.z + cc.y + cc.w) + b.z;
  v.w = 0.25f * (up.w + dn.w + cc.z + rt  ) + b.w;
  float4 o;
  o.x = (mw & 1u)        ? v.x : b.x;
  o.y = ((mw >> 1) & 1u) ? v.y : b.y;
  o.z = ((mw >> 2) & 1u) ? v.z : b.z;
  o.w = ((mw >> 3) & 1u) ? v.w : b.w;
  *(float4*)(Vout + c0) = o;
}

// --------------------------------------------------------- K5: finalize
__global__ void k_finalize(const float* __restrict__ V,
                           const unsigned* __restrict__ mask,
                           float* __restrict__ out) {
  int t = blockIdx.x * blockDim.x + threadIdx.x;   // 4 cells / thread
  int c0 = t << 2;
  float4 v = *(const float4*)(V + c0);
  unsigned mw = mask[c0 >> 5] >> (c0 & 31);
  // boundary cells already hold bc exactly; clamp applies only off-boundary
  float4 o;
  o.x = ((mw & 1u)        && v.x >= 1.0f) ? 0.95f : v.x;
  o.y = (((mw >> 1) & 1u) && v.y >= 1.0f) ? 0.95f : v.y;
  o.z = (((mw >> 2) & 1u) && v.z >= 1.0f) ? 0.95f : v.z;
  o.w = (((mw >> 3) & 1u) && v.w >= 1.0f) ? 0.95f : v.w;
  *(float4*)(out + c0) = o;
}

// ---------------------------------------------------------------------------
extern "C" void kernel_launch(void* const* d_in, const int* in_sizes, int n_in,
                              void* d_out, int out_size, void* d_ws, size_t ws_size,
                              hipStream_t stream) {
  const float* x  = (const float*)d_in[0];  // x
  const float* bt = (const float*)d_in[1];  // boundary_types (float 0/1/2)
  const float* bc = (const float*)d_in[2];  // boundary_conditions
  const float* C  = (const float*)d_in[3];  // C

  float* out  = (float*)d_out;              // tuple[0]: solved field
  float* cpos = out + (size_t)NCELLS;       // tuple[1]: C_pos

  // Workspace layout (needs 2*64MB + 2MB):
  float* bufA   = (float*)d_ws;             // V ping (holds V0, and V20 at end)
  float* Bfld   = bufA + (size_t)NCELLS;    // fused cost/boundary field
  unsigned* msk = (unsigned*)(Bfld + (size_t)NCELLS);  // interior bitmask

  // Reuse output halves as transient scratch (fully overwritten later):
  float* ob   = cpos;                       // objects_bounds (consumed by K2)
  float* H    = out;                        // horizontal conv result (consumed by K3)
  float* bufB = out;                        // V pong during iterations

  dim3 blk(256);
  dim3 grd1(NCELLS / 256);                  // 1 cell / thread
  dim3 grd4(NCELLS / (4 * 256));            // 4 cells / thread

  k_init<<<grd1, blk, 0, stream>>>(x, bt, bc, bufA, ob, msk);

  // 256x256 tiles, 1 wave per tile, 4 waves per block
  k_hconv_wmma<<<dim3((TILES_X * TILES_X) / 4), dim3(128), 0, stream>>>(ob, H);

  k_vconv_buildB<<<grd4, blk, 0, stream>>>(H, bt, bc, C, Bfld, cpos);

  // 20 Jacobi sweeps, ping-pong A<->B; even count ends in bufA
  for (int it = 0; it < 20; ++it) {
    const float* vin = (it & 1) ? bufB : bufA;
    float*       vout = (it & 1) ? bufA : bufB;
    k_jacobi<<<grd4, blk, 0, stream>>>(vin, Bfld, msk, vout);
  }

  k_finalize<<<grd4, blk, 0, stream>>>(bufA, msk, out);
}